// ForwardKinematicsLayer_49529562858096
// MI455X (gfx1250) — compile-verified
//
#include <hip/hip_runtime.h>
#include <stdint.h>

#define NJ 24
#define TPB 128
#define RSTRIDE 220   // padded dwords per element for rotations (216 data + 4 pad)
#define PSTRIDE 76    // padded dwords per element for positions (72 data + 4 pad)
#define ROT_CHUNKS 54 // 216 dwords / 4 per b128 chunk
#define POS_CHUNKS 18 // 72 dwords / 4

// ---- CDNA5 async global<->LDS engine (ASYNCcnt), GVS addressing: saddr + voffset + imm ----
// Rotation slab stage-in: copies 16-byte chunk c of the global slab at sbase to LDS, padding 16 bytes per 54 chunks
template <int I>
__device__ __forceinline__ void rot_load_all(unsigned lds_base, const float* sbase,
                                             unsigned voff, int tid) {
  if constexpr (I < ROT_CHUNKS) {
    const unsigned c   = (unsigned)(tid + TPB * I);
    const unsigned lds = lds_base + ((c + c / ROT_CHUNKS) << 4);
    asm volatile("global_load_async_to_lds_b128 %0, %1, %2 offset:%3 th:TH_LOAD_NT"
                 :: "v"(lds), "v"(voff), "s"(sbase), "n"(I * TPB * 16) : "memory");
    rot_load_all<I + 1>(lds_base, sbase, voff, tid);
  }
}

template <int I>
__device__ __forceinline__ void rot_store_all(unsigned lds_base, float* sbase,
                                              unsigned voff, int tid) {
  if constexpr (I < ROT_CHUNKS) {
    const unsigned c   = (unsigned)(tid + TPB * I);
    const unsigned lds = lds_base + ((c + c / ROT_CHUNKS) << 4);
    asm volatile("global_store_async_from_lds_b128 %0, %1, %2 offset:%3 th:TH_STORE_NT"
                 :: "v"(voff), "v"(lds), "s"(sbase), "n"(I * TPB * 16) : "memory");
    rot_store_all<I + 1>(lds_base, sbase, voff, tid);
  }
}

template <int I>
__device__ __forceinline__ void pos_store_all(unsigned pos_lds_base, float* sbase,
                                              unsigned voff, int tid) {
  if constexpr (I < POS_CHUNKS) {
    const unsigned c   = (unsigned)(tid + TPB * I);
    const unsigned lds = pos_lds_base + ((c + c / POS_CHUNKS) << 4);
    asm volatile("global_store_async_from_lds_b128 %0, %1, %2 offset:%3 th:TH_STORE_NT"
                 :: "v"(voff), "v"(lds), "s"(sbase), "n"(I * TPB * 16) : "memory");
    pos_store_all<I + 1>(pos_lds_base, sbase, voff, tid);
  }
}

__device__ __forceinline__ void wait_async0() {
  asm volatile("s_wait_asynccnt 0x0" ::: "memory");
}

__global__ void __launch_bounds__(TPB)
fk_kernel(const float* __restrict__ local_rots,
          const float* __restrict__ root_pos,
          const float* __restrict__ offsets,
          float* __restrict__ out_pos,
          float* __restrict__ out_rot) {
  __shared__ alignas(16) float smem[TPB * RSTRIDE + TPB * PSTRIDE];

  const int tid = threadIdx.x;
  // Low 32 bits of a generic pointer to LDS == workgroup-relative LDS byte offset
  const unsigned lds_base     = (unsigned)(uintptr_t)&smem[0];
  const unsigned pos_lds_base = lds_base + (unsigned)(TPB * RSTRIDE * 4);
  const unsigned voff         = (unsigned)tid << 4;  // per-lane 16B chunk offset

  const size_t blk = (size_t)blockIdx.x;

  // -------- Stage in: coalesced global -> padded LDS (async engine, NT) --------
  rot_load_all<0>(lds_base, local_rots + blk * (size_t)(TPB * 216), voff, tid);
  wait_async0();
  __syncthreads();

  // -------- FK chain, in place in this thread's LDS slab --------
  float* rl = &smem[tid * RSTRIDE];
  const int b = blockIdx.x * TPB + tid;

  float gp[NJ * 3];
  gp[0] = root_pos[(size_t)b * 3 + 0];
  gp[1] = root_pos[(size_t)b * 3 + 1];
  gp[2] = root_pos[(size_t)b * 3 + 2];

  const int par[NJ] = {-1,0,0,0,1,2,3,4,5,6,7,8,9,9,9,12,13,14,16,17,18,19,20,21};

#pragma unroll
  for (int j = 1; j < NJ; ++j) {
    const int p = par[j];
    float P[9], L[9], R[9];
#pragma unroll
    for (int q = 0; q < 9; ++q) P[q] = rl[p * 9 + q];   // parent global rot (already final)
#pragma unroll
    for (int q = 0; q < 9; ++q) L[q] = rl[j * 9 + q];   // local rot j
#pragma unroll
    for (int i = 0; i < 3; ++i)
#pragma unroll
      for (int k = 0; k < 3; ++k)
        R[i * 3 + k] = P[i * 3 + 0] * L[0 + k]
                     + P[i * 3 + 1] * L[3 + k]
                     + P[i * 3 + 2] * L[6 + k];
#pragma unroll
    for (int q = 0; q < 9; ++q) rl[j * 9 + q] = R[q];   // overwrite local with global

    const float ox = offsets[j * 3 + 0];  // uniform -> scalar loads
    const float oy = offsets[j * 3 + 1];
    const float oz = offsets[j * 3 + 2];
#pragma unroll
    for (int i = 0; i < 3; ++i)
      gp[j * 3 + i] = gp[p * 3 + i]
                    + P[i * 3 + 0] * ox + P[i * 3 + 1] * oy + P[i * 3 + 2] * oz;
  }

  // Positions: registers -> padded LDS staging region
  float* pl = &smem[TPB * RSTRIDE + tid * PSTRIDE];
#pragma unroll
  for (int q = 0; q < NJ * 3; ++q) pl[q] = gp[q];

  __syncthreads();

  // -------- Stage out: padded LDS -> coalesced global (async engine, NT) --------
  rot_store_all<0>(lds_base, out_rot + blk * (size_t)(TPB * 216), voff, tid);
  pos_store_all<0>(pos_lds_base, out_pos + blk * (size_t)(TPB * 72), voff, tid);
  wait_async0();  // belt-and-braces; S_ENDPGM also implies wait-idle
}

extern "C" void kernel_launch(void* const* d_in, const int* in_sizes, int n_in,
                              void* d_out, int out_size, void* d_ws, size_t ws_size,
                              hipStream_t stream) {
  const float* local_rots = (const float*)d_in[0]; // [B, 24, 3, 3]
  const float* root_pos   = (const float*)d_in[1]; // [B, 3]
  const float* offsets    = (const float*)d_in[2]; // [24, 3]
  float* out = (float*)d_out;

  const int B = in_sizes[1] / 3;              // 262144
  float* out_pos = out;                       // [B, 24, 3]
  float* out_rot = out + (size_t)B * 72;      // [B, 24, 3, 3]

  dim3 grid(B / TPB), block(TPB);             // B is an exact multiple of 128
  hipLaunchKernelGGL(fk_kernel, grid, block, 0, stream,
                     local_rots, root_pos, offsets, out_pos, out_rot);
}